// GCNEncoder_26654567039528
// MI455X (gfx1250) — compile-verified
//
#include <hip/hip_runtime.h>

typedef __attribute__((ext_vector_type(2))) float v2f;
typedef __attribute__((ext_vector_type(8))) float v8f;

// ---------------------------------------------------------------------------
// deg init / accumulate / rsqrt
// ---------------------------------------------------------------------------
__global__ void init_ones_kernel(float* __restrict__ p, int n) {
    int i = blockIdx.x * blockDim.x + threadIdx.x;
    if (i < n) p[i] = 1.0f;  // self-loop contributes 1 to every node's degree
}

__global__ void deg_accum_kernel(const int* __restrict__ dst, float* __restrict__ deg, int E) {
    int e = blockIdx.x * blockDim.x + threadIdx.x;
    if (e < E) atomicAdd(&deg[dst[e]], 1.0f);
}

__global__ void rsqrt_kernel(float* __restrict__ p, int n) {
    int i = blockIdx.x * blockDim.x + threadIdx.x;
    if (i < n) {
        float d = p[i];
        p[i] = (d > 0.0f) ? rsqrtf(d) : 0.0f;
    }
}

// ---------------------------------------------------------------------------
// fp32 WMMA GEMM: C[M,NCOL] = A[M,K] @ B[K,NCOL]   (row-major)
// K, NCOL compile-time so all inner-loop loads use immediate offsets.
// One wave32 per 16x16 output tile, V_WMMA_F32_16X16X4_F32 chained over K.
// ---------------------------------------------------------------------------
template <int K, int NCOL>
__global__ __launch_bounds__(128) void gemm_f32_wmma_kernel(
    const float* __restrict__ A, const float* __restrict__ B,
    float* __restrict__ C, int M) {
    const int lane    = threadIdx.x & 31;
    const int wave    = threadIdx.x >> 5;
    constexpr int tilesN = NCOL / 16;
    const int tileIdx = blockIdx.x * 4 + wave;
    const int nTiles  = ((M + 15) >> 4) * tilesN;
    if (tileIdx >= nTiles) return;  // wave-uniform: EXEC stays all-ones for WMMA

    const int tileM  = tileIdx / tilesN;
    const int tileN  = tileIdx - tileM * tilesN;
    const int half   = lane >> 4;   // 0: K pair {0,1}, 1: K pair {2,3}
    const int lane16 = lane & 15;

    int rowA = tileM * 16 + lane16;
    if (rowA >= M) rowA = M - 1;    // clamp loads; stores are bounds-checked
    const int colB = tileN * 16 + lane16;

    // Fixed base pointers; all loop loads are at compile-time immediates.
    const float* __restrict__ Abase = A + (size_t)rowA * K + half * 2;
    const float* __restrict__ Bbase = B + (size_t)(half * 2) * NCOL + colB;

    v8f acc = {};
#pragma unroll
    for (int k = 0; k < K; k += 4) {
        // A: 16x4 tile; lane holds row=lane16, two consecutive K values
        v2f a = *(const v2f*)(Abase + k);
        // B: 4x16 tile; lane holds col=lane16, same two K values
        v2f b;
        b.x = Bbase[k * NCOL];
        b.y = Bbase[k * NCOL + NCOL];
        acc = __builtin_amdgcn_wmma_f32_16x16x4_f32(
            false, a, false, b, (short)0, acc, false, false);
    }

    const int outCol = tileN * 16 + lane16;
    float* __restrict__ Crow = C + (size_t)(tileM * 16 + 8 * half) * NCOL + outCol;
#pragma unroll
    for (int v = 0; v < 8; ++v) {
        int outRow = tileM * 16 + v + 8 * half;  // C/D layout per ISA 7.12.2
        if (outRow < M) Crow[v * NCOL] = acc[v];
    }
}

// ---------------------------------------------------------------------------
// Edge scatter: out[dst] += h[src] * dinv[src]*dinv[dst]  (one wave per edge)
// ---------------------------------------------------------------------------
__global__ __launch_bounds__(256) void agg_edges_kernel(
    const float* __restrict__ h, const int* __restrict__ src,
    const int* __restrict__ dst, const float* __restrict__ dinv,
    float* __restrict__ out, int E, int D) {
    int e    = (blockIdx.x * blockDim.x + threadIdx.x) >> 5;
    int lane = threadIdx.x & 31;
    if (e >= E) return;
    int s = src[e];
    int d = dst[e];
    float nrm = dinv[s] * dinv[d];
    const float* __restrict__ hs = h + (size_t)s * D + lane;
    float* __restrict__ od = out + (size_t)d * D + lane;
#pragma unroll 4
    for (int f = 0; f < D; f += 32)
        atomicAdd(&od[f], hs[f] * nrm);
}

// ---------------------------------------------------------------------------
// Fused self-loop message + bias (+ optional ReLU):
//   out[i,f] += h[i,f]*dinv[i]^2 + b[f];  relu?
// ---------------------------------------------------------------------------
__global__ void finalize_kernel(const float* __restrict__ h,
                                const float* __restrict__ dinv,
                                const float* __restrict__ bias,
                                float* __restrict__ out,
                                int N, int D, int doRelu) {
    int i = blockIdx.x * blockDim.x + threadIdx.x;
    if (i >= N * D) return;
    int node = i / D;
    int f    = i - node * D;
    float di = dinv[node];
    float v  = out[i] + h[i] * di * di + bias[f];
    if (doRelu) v = fmaxf(v, 0.0f);
    out[i] = v;
}

// ---------------------------------------------------------------------------
// Global mean pool: per-graph sums + counts, then divide.
// ---------------------------------------------------------------------------
__global__ __launch_bounds__(256) void pool_accum_kernel(
    const float* __restrict__ h, const int* __restrict__ batch,
    float* __restrict__ sums, float* __restrict__ counts, int N, int D) {
    int node = (blockIdx.x * blockDim.x + threadIdx.x) >> 5;
    int lane = threadIdx.x & 31;
    if (node >= N) return;
    int g = batch[node];
    const float* __restrict__ hn = h + (size_t)node * D + lane;
    float* __restrict__ sg = sums + g * D + lane;
#pragma unroll 2
    for (int f = 0; f < D; f += 32)
        atomicAdd(&sg[f], hn[f]);
    if (lane == 0) atomicAdd(&counts[g], 1.0f);
}

__global__ void pool_div_kernel(const float* __restrict__ sums,
                                const float* __restrict__ counts,
                                float* __restrict__ out, int G, int D) {
    int i = blockIdx.x * blockDim.x + threadIdx.x;
    if (i >= G * D) return;
    int g = i / D;
    out[i] = sums[i] / fmaxf(counts[g], 1.0f);
}

// ---------------------------------------------------------------------------
// kernel_launch
// ---------------------------------------------------------------------------
extern "C" void kernel_launch(void* const* d_in, const int* in_sizes, int n_in,
                              void* d_out, int out_size, void* d_ws, size_t ws_size,
                              hipStream_t stream) {
    const float* x     = (const float*)d_in[0];
    const float* W1    = (const float*)d_in[1];
    const float* b1    = (const float*)d_in[2];
    const float* W2    = (const float*)d_in[3];
    const float* b2    = (const float*)d_in[4];
    const int*   ei    = (const int*)d_in[5];
    const int*   batch = (const int*)d_in[6];

    const int N    = in_sizes[6];           // 100000
    const int E    = in_sizes[5] / 2;       // 600000
    const int DH   = in_sizes[2];           // 128
    const int DOUT = in_sizes[4];           // 64
    const int G    = out_size / DOUT;       // 128

    const int* srcE = ei;
    const int* dstE = ei + E;

    // Workspace layout (floats):
    //   [0, N)                      : dinv  (deg -> deg^{-1/2})
    //   bufH = [N, N + N*DH)        : h1, later h2 (first N*DOUT), sums/counts after
    //   bufA = [.., .. + N*DH)      : a1 (post-agg layer1), later agg2 output
    float* ws     = (float*)d_ws;
    float* dinv   = ws;
    float* bufH   = dinv + N;
    float* bufA   = bufH + (size_t)N * DH;
    float* sums   = bufH + (size_t)N * DOUT;      // free space inside bufH region
    float* counts = sums + (size_t)G * DOUT;

    const int T = 256;

    // --- degree & normalization -------------------------------------------
    init_ones_kernel<<<(N + T - 1) / T, T, 0, stream>>>(dinv, N);
    deg_accum_kernel<<<(E + T - 1) / T, T, 0, stream>>>(dstE, dinv, E);
    rsqrt_kernel<<<(N + T - 1) / T, T, 0, stream>>>(dinv, N);

    // --- layer 1: h1 = x @ W1 (WMMA), aggregate, +b1, ReLU -----------------
    {
        int tiles = ((N + 15) / 16) * (128 / 16);
        gemm_f32_wmma_kernel<128, 128><<<(tiles + 3) / 4, 128, 0, stream>>>(x, W1, bufH, N);
    }
    hipMemsetAsync(bufA, 0, (size_t)N * DH * sizeof(float), stream);
    agg_edges_kernel<<<(E + 7) / 8, T, 0, stream>>>(bufH, srcE, dstE, dinv, bufA, E, DH);
    finalize_kernel<<<((N * DH) + T - 1) / T, T, 0, stream>>>(bufH, dinv, b1, bufA, N, DH, 1);

    // --- layer 2: h2 = a1 @ W2 (WMMA), aggregate, +b2 ----------------------
    {
        int tiles = ((N + 15) / 16) * (64 / 16);
        gemm_f32_wmma_kernel<128, 64><<<(tiles + 3) / 4, 128, 0, stream>>>(bufA, W2, bufH, N);
    }
    hipMemsetAsync(bufA, 0, (size_t)N * DOUT * sizeof(float), stream);
    agg_edges_kernel<<<(E + 7) / 8, T, 0, stream>>>(bufH, srcE, dstE, dinv, bufA, E, DOUT);
    finalize_kernel<<<((N * DOUT) + T - 1) / T, T, 0, stream>>>(bufH, dinv, b2, bufA, N, DOUT, 0);

    // --- global mean pool --------------------------------------------------
    hipMemsetAsync(sums, 0, (size_t)(G * DOUT + G) * sizeof(float), stream);
    pool_accum_kernel<<<(N + 7) / 8, T, 0, stream>>>(bufA, batch, sums, counts, N, DOUT);
    pool_div_kernel<<<(G * DOUT + T - 1) / T, T, 0, stream>>>(sums, counts, (float*)d_out, G, DOUT);
}